// VarianceAdaptor_27023934227161
// MI455X (gfx1250) — compile-verified
//
#include <hip/hip_runtime.h>
#include <hip/hip_bf16.h>

typedef float v2f __attribute__((ext_vector_type(2)));
typedef float v4f __attribute__((ext_vector_type(4)));
typedef float v8f __attribute__((ext_vector_type(8)));

#define B_     32
#define T_IN_  512
#define T_OUT_ 2048
#define ADIM_  384

// ---------------------------------------------------------------------------
// Kernel 1: per-batch masked-duration scan + searchsorted -> source row index
// idxbuf[b*T_OUT + t] = source phone row (clipped), or -1 if t >= total frames
// ---------------------------------------------------------------------------
__global__ __launch_bounds__(T_IN_) void lr_index_kernel(
    const int* __restrict__ durations,
    const int* __restrict__ input_lengths,
    int* __restrict__ idxbuf)
{
    __shared__ int cum[T_IN_];
    const int b = blockIdx.x;
    const int i = threadIdx.x;
    const int len = input_lengths[b];

    int d = (i < len) ? durations[b * T_IN_ + i] : 0;

    // torch rule: if masked total == 0, fall back to ones inside the mask
    for (int pass = 0; pass < 2; ++pass) {
        cum[i] = d;
        __syncthreads();
        // Hillis-Steele inclusive scan
        for (int off = 1; off < T_IN_; off <<= 1) {
            int v   = cum[i];
            int add = (i >= off) ? cum[i - off] : 0;
            __syncthreads();
            cum[i] = v + add;
            __syncthreads();
        }
        int total = cum[T_IN_ - 1];   // uniform
        __syncthreads();
        if (pass == 0) {
            if (total != 0) break;    // uniform branch
            d = (i < len) ? 1 : 0;
        }
    }

    const int total = cum[T_IN_ - 1];

    for (int t = i; t < T_OUT_; t += T_IN_) {
        // searchsorted(cum, t, side='right') == count of cum entries <= t
        int lo = 0, hi = T_IN_;
        while (lo < hi) {
            int mid = (lo + hi) >> 1;
            if (cum[mid] <= t) lo = mid + 1; else hi = mid;
        }
        int idx = (lo > T_IN_ - 1) ? (T_IN_ - 1) : lo;
        idxbuf[b * T_OUT_ + t] = (t < total) ? idx : -1;
    }
}

// ---------------------------------------------------------------------------
// Kernel 2: fused length-regulate gather + pitch/energy embed via WMMA f32 K=4
// One wave owns a 16(t) x 16(a) output tile.
//   D = A_pitch(16x4) * B_pitch(4x16)        (4 pitch dims, K exactly 4)
//     + A_misc (16x4) * B_misc (4x16)        (K0 = energy, K1 = bias, K2,3 = 0)
//     + C = gathered hs tile (zero-masked)
// ---------------------------------------------------------------------------
__global__ __launch_bounds__(256) void va_fused_wmma_kernel(
    const float* __restrict__ hs,
    const float* __restrict__ pt,
    const float* __restrict__ et,
    const float* __restrict__ pw,
    const float* __restrict__ pb,
    const float* __restrict__ ew,
    const float* __restrict__ eb,
    const int*  __restrict__ idxbuf,
    float* __restrict__ out)
{
    const int lane = threadIdx.x & 31;
    const int wave = threadIdx.x >> 5;          // 8 waves -> 8 a-tiles per block
    const int t0   = blockIdx.x * 16;
    const int a0   = (blockIdx.y * 8 + wave) * 16;
    const int b    = blockIdx.z;
    const int half = lane >> 4;                 // 0: K=0,1  1: K=2,3 (A/B layout)
    const int l    = lane & 15;                 // M (for A) / N (for B,C,D)

    // ---- A operand: pitch weights, A[m][k] = pw[a0+m][k] ----
    const int ch = a0 + l;
    v2f a_pitch;
    {
        const float* p = pw + ch * 4 + half * 2;   // row-major [384,4]
        a_pitch.x = p[0];
        a_pitch.y = p[1];
    }

    // ---- A operand 2: K0 = energy weight, K1 = combined bias, K2/K3 = 0 ----
    v2f a_misc;
    a_misc.x = half ? 0.0f : ew[ch];
    a_misc.y = half ? 0.0f : (pb[ch] + eb[ch]);

    // ---- B operand: pitch frames, B[k][n] = pt[b, t0+n, k] ----
    const int  t     = t0 + l;
    const long frame = (long)b * T_OUT_ + t;
    v2f b_pitch;
    {
        const float* p = pt + frame * 4 + half * 2; // row-major [B,2048,4]
        b_pitch.x = p[0];
        b_pitch.y = p[1];
    }

    // ---- B operand 2: K0 = energy frame, K1 = 1.0 (bias lane), K2/K3 = 0 ----
    v2f b_misc;
    b_misc.x = half ? 0.0f : et[frame];
    b_misc.y = half ? 0.0f : 1.0f;

    // ---- C init: gathered hs tile, branchless mask keeps EXEC all-ones ----
    const int   src = idxbuf[frame];
    const int   row = (src < 0) ? 0 : src;
    const float msk = (src < 0) ? 0.0f : 1.0f;
    const float* hp = hs + ((long)b * T_IN_ + row) * ADIM_ + a0 + half * 8;
    v4f h0 = *(const v4f*)hp;
    v4f h1 = *(const v4f*)(hp + 4);
    v8f c;
    c[0] = h0.x * msk; c[1] = h0.y * msk; c[2] = h0.z * msk; c[3] = h0.w * msk;
    c[4] = h1.x * msk; c[5] = h1.y * msk; c[6] = h1.z * msk; c[7] = h1.w * msk;

    // ---- two chained f32 WMMAs: D = Am*Bm + (Ap*Bp + C) ----
    c = __builtin_amdgcn_wmma_f32_16x16x4_f32(false, a_pitch, false, b_pitch,
                                              (short)0, c, false, false);
    c = __builtin_amdgcn_wmma_f32_16x16x4_f32(false, a_misc, false, b_misc,
                                              (short)0, c, false, false);

    // ---- store: C layout -> lane owns 8 consecutive channels of row t0+l ----
    float* op = out + frame * ADIM_ + a0 + half * 8;
    *(v4f*)op       = (v4f){c[0], c[1], c[2], c[3]};
    *(v4f*)(op + 4) = (v4f){c[4], c[5], c[6], c[7]};
}

extern "C" void kernel_launch(void* const* d_in, const int* in_sizes, int n_in,
                              void* d_out, int out_size, void* d_ws, size_t ws_size,
                              hipStream_t stream)
{
    (void)in_sizes; (void)n_in; (void)out_size; (void)ws_size;

    const float* hs            = (const float*)d_in[0];
    const int*   durations     = (const int*)d_in[1];
    const int*   input_lengths = (const int*)d_in[2];
    const float* pitch_target  = (const float*)d_in[3];
    const float* energy_target = (const float*)d_in[4];
    // d_in[5] duration_mask, d_in[6] variance_mask: unused by the reference math
    const float* pitch_w       = (const float*)d_in[7];
    const float* pitch_b       = (const float*)d_in[8];
    const float* energy_w      = (const float*)d_in[9];
    const float* energy_b      = (const float*)d_in[10];

    float* out    = (float*)d_out;
    int*   idxbuf = (int*)d_ws;            // B * T_OUT ints = 256 KB scratch

    lr_index_kernel<<<B_, T_IN_, 0, stream>>>(durations, input_lengths, idxbuf);

    dim3 grid(T_OUT_ / 16, ADIM_ / (16 * 8), B_);   // (128, 3, 32)
    va_fused_wmma_kernel<<<grid, 256, 0, stream>>>(
        hs, pitch_target, energy_target,
        pitch_w, pitch_b, energy_w, energy_b,
        idxbuf, out);
}